// TopKGate_83554293776536
// MI455X (gfx1250) — compile-verified
//
#include <hip/hip_runtime.h>
#include <hip/hip_bf16.h>

typedef __attribute__((ext_vector_type(2))) float v2f;
typedef __attribute__((ext_vector_type(8))) float v8f;

#define S_TOK 16384
#define DIM   4096
#define NEXP  64
#define CAP   256
#define KC    256            // K-chunk staged in LDS
#define ROWS_PER_BLOCK 128   // 8 waves * 16 rows

// -------- init: me accumulator = 0, dispatch slots = -1 --------
__global__ void moe_init(float* meAcc, int* dispatch) {
    int i = blockIdx.x * 256 + threadIdx.x;
    if (i < NEXP) meAcc[i] = 0.0f;
    if (i < NEXP * CAP) dispatch[i] = -1;
}

// -------- K1: router GEMM (WMMA f32 16x16x4) + softmax + argmax --------
// dyn LDS: wgTile[KC*64] (64KB, B-fragment order) | logits[128*65] (33280B) | meLds[64]
__global__ void __launch_bounds__(256)
moe_gate_gemm(const float* __restrict__ x, const float* __restrict__ wg,
              int* __restrict__ expIdx, float* __restrict__ gateVal,
              float* __restrict__ meAcc) {
    extern __shared__ char smem[];
    float* wgTile = (float*)smem;                                  // KC*64 floats
    float* logits = (float*)(smem + KC * NEXP * 4);                // 128*65 floats
    float* meLds  = (float*)(smem + KC * NEXP * 4 + 128 * 65 * 4); // 64 floats

    const int tid  = threadIdx.x;
    const int wave = tid >> 5;
    const int lane = tid & 31;
    const int col  = lane & 15;   // N / M position within half-wave
    const int hi   = lane >> 4;   // K half selector for A and B fragments
    const int rowBase = blockIdx.x * ROWS_PER_BLOCK + wave * 16;

    if (tid < NEXP) meLds[tid] = 0.0f;

    v8f acc0 = {}, acc1 = {}, acc2 = {}, acc3 = {};
    const float* aPtr = x + (size_t)(rowBase + col) * DIM + hi * 2;

    for (int k0 = 0; k0 < DIM; k0 += KC) {
        __syncthreads();
        // stage wg[k0..k0+KC) into LDS in WMMA B-fragment order:
        //   idx = ((g*2+hi)*4 + n)*32 + col*2 + kp
        //   with k = k0 + 4g + 2hi + kp, e = 16n + col
        {
            const float4* src = (const float4*)(wg + (size_t)k0 * NEXP);
#pragma unroll
            for (int j = 0; j < (KC * NEXP / 4) / 256; ++j) {
                int f = j * 256 + tid;          // float4 index (coalesced global read)
                float4 v = src[f];
                int kq  = (f * 4) >> 6;         // k - k0 (row of wg chunk)
                int e0  = (f * 4) & 63;         // first of 4 consecutive experts
                int g   = kq >> 2;
                int hh  = (kq >> 1) & 1;
                int kp  = kq & 1;
                int n   = e0 >> 4;              // e0 multiple of 4 -> same n-tile for all 4
                int c0  = e0 & 15;
                float* dstp = wgTile + (((g * 2 + hh) * 4 + n) * 32 + c0 * 2 + kp);
                dstp[0] = v.x; dstp[2] = v.y; dstp[4] = v.z; dstp[6] = v.w;
            }
        }
        if (k0 + KC < DIM)
            __builtin_prefetch(aPtr + k0 + KC, 0, 0); // global_prefetch next A chunk
        __syncthreads();

#pragma unroll 4
        for (int kk = 0; kk < KC; kk += 4) {
            // A fragment: 16x4 f32, lanes 0-15 hold K=k,k+1; lanes 16-31 hold K=k+2,k+3
            v2f a = *(const v2f*)(aPtr + k0 + kk);
            // B fragments: one aligned ds_load_b64 per 16-expert tile
            const float* bp = wgTile + (((kk >> 2) * 2 + hi) * 128 + col * 2);
            v2f b0 = *(const v2f*)(bp + 0);
            v2f b1 = *(const v2f*)(bp + 32);
            v2f b2 = *(const v2f*)(bp + 64);
            v2f b3 = *(const v2f*)(bp + 96);
            acc0 = __builtin_amdgcn_wmma_f32_16x16x4_f32(false, a, false, b0, (short)0, acc0, false, false);
            acc1 = __builtin_amdgcn_wmma_f32_16x16x4_f32(false, a, false, b1, (short)0, acc1, false, false);
            acc2 = __builtin_amdgcn_wmma_f32_16x16x4_f32(false, a, false, b2, (short)0, acc2, false, false);
            acc3 = __builtin_amdgcn_wmma_f32_16x16x4_f32(false, a, false, b3, (short)0, acc3, false, false);
        }
    }

    // spill C tiles: VGPR r holds row (r + hi*8), N = col (+16n per tile)
    const int lrow = wave * 16 + hi * 8;
#pragma unroll
    for (int r = 0; r < 8; ++r) {
        float* lr = logits + (size_t)(lrow + r) * 65 + col;
        lr[0]  = acc0[r];
        lr[16] = acc1[r];
        lr[32] = acc2[r];
        lr[48] = acc3[r];
    }
    __syncthreads();

    // one thread per row: argmax + softmax (padded stride 65 -> conflict-free)
    if (tid < ROWS_PER_BLOCK) {
        const float* row = logits + (size_t)tid * 65;
        float best = row[0];
        int   bi   = 0;
        for (int e = 1; e < NEXP; ++e) {
            float v = row[e];
            if (v > best) { best = v; bi = e; }   // first-occurrence argmax
        }
        float denom = 0.0f;
        for (int e = 0; e < NEXP; ++e) denom += __expf(row[e] - best);
        float inv = 1.0f / denom;                 // gate prob of the argmax expert
        for (int e = 0; e < NEXP; ++e)
            atomicAdd(&meLds[e], __expf(row[e] - best) * inv);
        int s = blockIdx.x * ROWS_PER_BLOCK + tid;
        expIdx[s]  = bi;
        gateVal[s] = inv;
    }
    __syncthreads();
    if (tid < NEXP) atomicAdd(&meAcc[tid], meLds[tid]);
}

// -------- K2: per-block (256 tokens, token-ordered) expert histogram --------
__global__ void moe_hist(const int* __restrict__ expIdx, int* __restrict__ counts) {
    __shared__ int h[NEXP];
    int tid = threadIdx.x;
    if (tid < NEXP) h[tid] = 0;
    __syncthreads();
    int s = blockIdx.x * 256 + tid;
    atomicAdd(&h[expIdx[s]], 1);
    __syncthreads();
    if (tid < NEXP) counts[blockIdx.x * NEXP + tid] = h[tid];
}

// -------- K3: per-expert exclusive scan over blocks + aux loss --------
__global__ void moe_scan_loss(const int* __restrict__ counts, int* __restrict__ base,
                              const float* __restrict__ meAcc, float* __restrict__ lossOut) {
    __shared__ float red[NEXP];
    int e = threadIdx.x;             // 64 threads
    int run = 0;
    for (int b = 0; b < S_TOK / 256; ++b) {
        base[b * NEXP + e] = run;
        run += counts[b * NEXP + e];
    }
    red[e] = meAcc[e] * (float)run;  // (me*S) * (ce*S)
    __syncthreads();
    if (e == 0) {
        float t = 0.0f;
        for (int i = 0; i < NEXP; ++i) t += red[i];
        lossOut[0] = 0.01f * (float)NEXP * t / ((float)S_TOK * (float)S_TOK);
    }
}

// -------- K4: token-ordered placement, capacity cut, scatter --------
__global__ void __launch_bounds__(256)
moe_place(const int* __restrict__ expIdx, const float* __restrict__ gateVal,
          const int* __restrict__ base, int* __restrict__ dispatch,
          float* __restrict__ gatesOut) {
    __shared__ int wc[8 * NEXP];
    int tid = threadIdx.x, wave = tid >> 5, lane = tid & 31;
    for (int i = tid; i < 8 * NEXP; i += 256) wc[i] = 0;

    int s = blockIdx.x * 256 + tid;
    int e = expIdx[s];

    // intra-wave ordered rank among lanes with the same expert (6 ballots)
    unsigned m = 0xFFFFFFFFu;
#pragma unroll
    for (int bit = 0; bit < 6; ++bit) {
        unsigned b = __builtin_amdgcn_ballot_w32((bool)((e >> bit) & 1));
        m &= ((e >> bit) & 1) ? b : ~b;
    }
    int rank = __popc(m & ((1u << lane) - 1u));

    __syncthreads();
    if (rank == 0) wc[wave * NEXP + e] = __popc(m);   // leader writes wave count
    __syncthreads();

    int off = base[blockIdx.x * NEXP + e];
    for (int w = 0; w < wave; ++w) off += wc[w * NEXP + e];
    int loc = off + rank;

    float g = gateVal[s];
    if (loc < CAP) {
        dispatch[e * CAP + loc] = s;
        gatesOut[s] = g;
    } else {
        gatesOut[s] = 0.0f;   // dropped token
    }
}

extern "C" void kernel_launch(void* const* d_in, const int* in_sizes, int n_in,
                              void* d_out, int out_size, void* d_ws, size_t ws_size,
                              hipStream_t stream) {
    (void)in_sizes; (void)n_in; (void)out_size; (void)ws_size;
    const float* x  = (const float*)d_in[0];   // [16384, 4096] f32
    const float* wg = (const float*)d_in[1];   // [4096, 64]    f32

    // workspace partition
    char* ws = (char*)d_ws;
    int*   expIdx  = (int*)ws;                                   // S ints
    float* gateVal = (float*)(ws + (size_t)S_TOK * 4);           // S floats
    float* meAcc   = (float*)(ws + (size_t)S_TOK * 8);           // 64 floats
    int*   counts  = (int*)(ws + (size_t)S_TOK * 8 + 256);       // 64*64 ints
    int*   baseOff = (int*)(ws + (size_t)S_TOK * 8 + 256 + 64 * NEXP * 4);

    // output: [loss(1) | gates(S) | dispatch(E*CAP) as int32 bits]
    float* out      = (float*)d_out;
    float* lossOut  = out;
    float* gatesOut = out + 1;
    int*   dispatch = (int*)(out + 1 + S_TOK);

    moe_init<<<NEXP * CAP / 256, 256, 0, stream>>>(meAcc, dispatch);

    size_t smem = (size_t)KC * NEXP * 4 + 128 * 65 * 4 + NEXP * 4; // ~99 KB (<320KB WGP LDS)
    moe_gate_gemm<<<S_TOK / ROWS_PER_BLOCK, 256, smem, stream>>>(x, wg, expIdx, gateVal, meAcc);

    moe_hist<<<S_TOK / 256, 256, 0, stream>>>(expIdx, counts);
    moe_scan_loss<<<1, NEXP, 0, stream>>>(counts, baseOff, meAcc, lossOut);
    moe_place<<<S_TOK / 256, 256, 0, stream>>>(expIdx, gateVal, baseOff, dispatch, gatesOut);
}